// GCNConv_74603581931522
// MI455X (gfx1250) — compile-verified
//
#include <hip/hip_runtime.h>
#include <hip/hip_bf16.h>
#include <stdint.h>

// ---------------------------------------------------------------------------
// GCNConv on MI455X (gfx1250, wave32)
//   deg -> dinv -> self-loop init of agg -> edge scatter -> WMMA fp32 GEMM
// ---------------------------------------------------------------------------

typedef __attribute__((ext_vector_type(2))) float v2f;
typedef __attribute__((ext_vector_type(8))) float v8f;

#define C_DIM 64

// ---- 1) deg[i] = 1.0 (self loop contributes weight 1) ----------------------
__global__ void gcn_init_deg(float* __restrict__ deg, int n) {
    int i = blockIdx.x * blockDim.x + threadIdx.x;
    if (i < n) deg[i] = 1.0f;
}

// ---- 2) deg[row[e]] += ew[e] ----------------------------------------------
__global__ void gcn_deg_scatter(const long long* __restrict__ rows,
                                const float* __restrict__ ew,
                                float* __restrict__ deg, long long E) {
    long long e = (long long)blockIdx.x * blockDim.x + threadIdx.x;
    if (e < E) {
        long long r = rows[e];
        __hip_atomic_fetch_add(&deg[r], ew[e],
                               __ATOMIC_RELAXED, __HIP_MEMORY_SCOPE_AGENT);
    }
}

// ---- 3) dinv = rsqrt(deg) --------------------------------------------------
__global__ void gcn_dinv(const float* __restrict__ deg,
                         float* __restrict__ dinv, int n) {
    int i = blockIdx.x * blockDim.x + threadIdx.x;
    if (i < n) {
        float d = deg[i];
        dinv[i] = (d > 0.0f) ? __frsqrt_rn(d) : 0.0f;
    }
}

// ---- 4) agg[i][c] = dinv[i]^2 * x[i][c]  (self loop; initializes agg) ------
__global__ void gcn_self_msg(const float* __restrict__ x,
                             const float* __restrict__ dinv,
                             float* __restrict__ agg, long long total) {
    long long idx = (long long)blockIdx.x * blockDim.x + threadIdx.x;
    if (idx < total) {
        long long node = idx >> 6;  // /64
        float di = dinv[node];
        agg[idx] = di * di * x[idx];
    }
}

// ---- 5) one wave (32 lanes) per edge: agg[row] += w * x[col] ---------------
// lane L handles channels L and L+32. row/col/ew are wave-uniform -> scalar.
__global__ void gcn_edge_scatter(const long long* __restrict__ rows,
                                 const long long* __restrict__ cols,
                                 const float* __restrict__ ew,
                                 const float* __restrict__ dinv,
                                 const float* __restrict__ x,
                                 float* __restrict__ agg, long long E) {
    long long e = (long long)blockIdx.x * (blockDim.x >> 5) + (threadIdx.x >> 5);
    if (e >= E) return;
    int lane = threadIdx.x & 31;
    long long r = rows[e];
    long long c = cols[e];
    float w = dinv[r] * ew[e] * dinv[c];
    const float* xs = x + c * C_DIM;
    float* ad = agg + r * C_DIM;
    float m0 = xs[lane] * w;
    float m1 = xs[lane + 32] * w;
    __hip_atomic_fetch_add(ad + lane, m0,
                           __ATOMIC_RELAXED, __HIP_MEMORY_SCOPE_AGENT);
    __hip_atomic_fetch_add(ad + lane + 32, m1,
                           __ATOMIC_RELAXED, __HIP_MEMORY_SCOPE_AGENT);
}

// ---- 6) out = agg @ W + bias via V_WMMA_F32_16X16X4_F32 --------------------
// Block: 128 threads = 4 waves; wave w computes 16x16 tile at cols [16w,16w+16)
// of output rows [16*blockIdx.x, +16). K=64 accumulated in 16 steps of 4.
//
// fp32 WMMA VGPR layouts (wave32):
//  A 16x4 : lane l (m = l&15, h = l>>4) holds A[m][2h+0], A[m][2h+1] in v0,v1
//  B 4x16 : lane l (n = l&15, h = l>>4) holds B[2h+0][n], B[2h+1][n] in v0,v1
//  C 16x16: vgpr r, lane l -> C[r + 8*(l>>4)][l&15]
__global__ void gcn_gemm_wmma(const float* __restrict__ agg,
                              const float* __restrict__ W,
                              const float* __restrict__ bias,
                              float* __restrict__ out, int nrows) {
    const int wave = threadIdx.x >> 5;   // 0..3 -> output col tile
    const int lane = threadIdx.x & 31;
    const int lm   = lane & 15;
    const int lh   = lane >> 4;          // 0 or 1
    const int m0   = blockIdx.x * 16;
    const int n0   = wave * 16;

    v8f c = {};
    const float* arow = agg + (size_t)(m0 + lm) * C_DIM + lh * 2;
    const float* bcol = W + (size_t)(lh * 2) * C_DIM + n0 + lm;

#pragma unroll
    for (int k = 0; k < C_DIM; k += 4) {
        v2f a = *(const v2f*)(arow + k);         // A[m][k+2h], A[m][k+2h+1]
        v2f b;
        b.x = bcol[(size_t)k * C_DIM];           // B[k+2h+0][n]
        b.y = bcol[(size_t)k * C_DIM + C_DIM];   // B[k+2h+1][n]
        c = __builtin_amdgcn_wmma_f32_16x16x4_f32(
                /*neg_a=*/false, a, /*neg_b=*/false, b,
                /*c_mod=*/(short)0, c, /*reuse_a=*/false, /*reuse_b=*/false);
    }

    float bv = bias[n0 + lm];
#pragma unroll
    for (int r = 0; r < 8; ++r) {
        int m = m0 + r + lh * 8;
        if (m < nrows) out[(size_t)m * C_DIM + n0 + lm] = c[r] + bv;
    }
}

// ---------------------------------------------------------------------------
extern "C" void kernel_launch(void* const* d_in, const int* in_sizes, int n_in,
                              void* d_out, int out_size, void* d_ws, size_t ws_size,
                              hipStream_t stream) {
    const float*      x    = (const float*)d_in[0];       // [N, 64]
    const long long*  ei   = (const long long*)d_in[1];   // [2, E] int64
    const float*      ew   = (const float*)d_in[2];       // [E]
    const float*      W    = (const float*)d_in[3];       // [64, 64]
    const float*      bias = (const float*)d_in[4];       // [64]
    float*            out  = (float*)d_out;               // [N, 64]

    const int       N = in_sizes[0] / C_DIM;
    const long long E = (long long)in_sizes[2];
    const long long* rows = ei;
    const long long* cols = ei + E;

    // workspace layout: deg[N] | dinv[N] | agg[N*64]   (all 16B aligned)
    float* deg  = (float*)d_ws;
    float* dinv = deg + N;
    float* agg  = dinv + N;

    const int TPB = 256;

    // 1) deg = 1.0
    gcn_init_deg<<<(N + TPB - 1) / TPB, TPB, 0, stream>>>(deg, N);

    // 2) deg[row] += ew
    gcn_deg_scatter<<<(unsigned)((E + TPB - 1) / TPB), TPB, 0, stream>>>(
        rows, ew, deg, E);

    // 3) dinv = rsqrt(deg)
    gcn_dinv<<<(N + TPB - 1) / TPB, TPB, 0, stream>>>(deg, dinv, N);

    // 4) agg = dinv^2 * x  (self loops; also initializes agg)
    long long total = (long long)N * C_DIM;
    gcn_self_msg<<<(unsigned)((total + TPB - 1) / TPB), TPB, 0, stream>>>(
        x, dinv, agg, total);

    // 5) edge scatter: one wave per edge (8 waves / 256-thread block)
    long long wpb = TPB / 32;
    gcn_edge_scatter<<<(unsigned)((E + wpb - 1) / wpb), TPB, 0, stream>>>(
        rows, cols, ew, dinv, x, agg, E);

    // 6) out = agg @ W + bias  (fp32 WMMA, 16-row tiles, 4 col-tile waves)
    int row_tiles = (N + 15) / 16;
    gcn_gemm_wmma<<<row_tiles, 128, 0, stream>>>(agg, W, bias, out, N);
}